// SAGE_70875550319063
// MI455X (gfx1250) — compile-verified
//
#include <hip/hip_runtime.h>
#include <hip/hip_bf16.h>
#include <math.h>

typedef __attribute__((ext_vector_type(2))) float v2f;
typedef __attribute__((ext_vector_type(8))) float v8f;

#define D 128
#define ROW_TILES 4   // row tiles (16 rows each) per wave -> 64 rows per block

// ---------------------------------------------------------------------------
// Degree: deg[v] = number of incoming edges (edge_dst == v)
// ---------------------------------------------------------------------------
__global__ void deg_kernel(const int* __restrict__ edst, float* __restrict__ deg, int E) {
    int i = blockIdx.x * blockDim.x + threadIdx.x;
    if (i < E) {
        unsafeAtomicAdd(&deg[edst[i]], 1.0f);
    }
}

// ---------------------------------------------------------------------------
// Scatter-add: agg[edge_dst[e]][:] += X[edge_src[e]][:]
// One wave (32 lanes) per edge; each lane handles 4 contiguous dims (float4).
// ---------------------------------------------------------------------------
__global__ void scatter_kernel(const float* __restrict__ X,
                               const int* __restrict__ esrc,
                               const int* __restrict__ edst,
                               float* __restrict__ agg, int E) {
    int gid  = blockIdx.x * blockDim.x + threadIdx.x;
    int e    = gid >> 5;
    int lane = gid & 31;
    if (e >= E) return;
    int s = esrc[e];
    int d = edst[e];
    const float4 v = ((const float4*)(X + (size_t)s * D))[lane];
    float* p = agg + (size_t)d * D + lane * 4;
    unsafeAtomicAdd(p + 0, v.x);
    unsafeAtomicAdd(p + 1, v.y);
    unsafeAtomicAdd(p + 2, v.z);
    unsafeAtomicAdd(p + 3, v.w);
}

// ---------------------------------------------------------------------------
// SAGE layer: out = X @ Wself + (agg/max(deg,1)) @ Wneigh + b   [+ ReLU]
//
// V_WMMA_F32_16X16X4_F32, one wave per 64x16 output slab (4 row tiles that
// share each B load). K = 128 -> 32 k-steps, 8 chained WMMAs per step.
//
// A 16x4 f32 layout (2 VGPRs/lane): lanes 0-15 hold (K, K+1) of row M=lane;
// lanes 16-31 hold (K+2, K+3) of row M=lane-16.
// B 4x16 f32 layout mirrors A: lanes 0-15 hold rows (K, K+1) of col N=lane;
// lanes 16-31 hold rows (K+2, K+3) of col N=lane-16.
// C/D 16x16 f32 (8 VGPRs): VGPR g -> row g (lanes 0-15, N=lane) and
// row g+8 (lanes 16-31, N=lane-16).
// ---------------------------------------------------------------------------
__global__ void __launch_bounds__(256)
sage_layer_kernel(const float* __restrict__ X,
                  const float* __restrict__ agg,
                  const float* __restrict__ deg,
                  const float* __restrict__ Wself,
                  const float* __restrict__ Wneigh,
                  const float* __restrict__ bias,
                  float* __restrict__ out,
                  int relu) {
    const int lane = threadIdx.x & 31;
    const int wave = threadIdx.x >> 5;            // 0..7 : column tile
    const int row0 = blockIdx.x * (16 * ROW_TILES);
    const int col0 = wave * 16;
    const int l15  = lane & 15;
    const int hi   = (lane >= 16) ? 1 : 0;
    const int ka   = hi ? 2 : 0;                  // K sub-offset for this lane half

    // Per-lane fixed base row (tile t adds t*16)
    const float* xrow = X   + (size_t)(row0 + l15) * D;
    const float* hrow = agg + (size_t)(row0 + l15) * D;

    float invdeg[ROW_TILES];
#pragma unroll
    for (int t = 0; t < ROW_TILES; ++t)
        invdeg[t] = 1.0f / fmaxf(deg[row0 + t * 16 + l15], 1.0f);

    v8f acc[ROW_TILES];
#pragma unroll
    for (int t = 0; t < ROW_TILES; ++t) acc[t] = (v8f){};

    for (int k = 0; k < D; k += 4) {
        const int kk = k + ka;

        // B operands: shared across all 4 row tiles
        const float* wS = Wself  + (size_t)kk * D + col0 + l15;
        const float* wN = Wneigh + (size_t)kk * D + col0 + l15;
        v2f bS, bN;
        bS.x = wS[0];  bS.y = wS[D];
        bN.x = wN[0];  bN.y = wN[D];

#pragma unroll
        for (int t = 0; t < ROW_TILES; ++t) {
            const int off = t * 16 * D + kk;
            v2f aS, aN;
            aS.x = xrow[off + 0];
            aS.y = xrow[off + 1];
            aN.x = hrow[off + 0] * invdeg[t];
            aN.y = hrow[off + 1] * invdeg[t];

            acc[t] = __builtin_amdgcn_wmma_f32_16x16x4_f32(
                false, aS, false, bS, (short)0, acc[t], false, false);
            acc[t] = __builtin_amdgcn_wmma_f32_16x16x4_f32(
                false, aN, false, bN, (short)0, acc[t], false, false);
        }
    }

    const int c = col0 + l15;
    const float bc = bias[c];
#pragma unroll
    for (int t = 0; t < ROW_TILES; ++t) {
#pragma unroll
        for (int g = 0; g < 8; ++g) {
            int r = row0 + t * 16 + g + (hi ? 8 : 0);
            float v = acc[t][g] + bc;
            if (relu) v = fmaxf(v, 0.0f);
            out[(size_t)r * D + c] = v;
        }
    }
}

// ---------------------------------------------------------------------------
// Edge score: out[e] = sigmoid( dot(h[src[e]], h[dst[e]]) )
// One wave per edge, 4 dims/lane, wave32 shuffle reduction.
// ---------------------------------------------------------------------------
__global__ void edge_score_kernel(const float* __restrict__ h,
                                  const int* __restrict__ src,
                                  const int* __restrict__ dst,
                                  float* __restrict__ out, int E) {
    int gid  = blockIdx.x * blockDim.x + threadIdx.x;
    int e    = gid >> 5;
    int lane = gid & 31;
    if (e >= E) return;
    const float4 a = ((const float4*)(h + (size_t)src[e] * D))[lane];
    const float4 b = ((const float4*)(h + (size_t)dst[e] * D))[lane];
    float s = a.x * b.x + a.y * b.y + a.z * b.z + a.w * b.w;
#pragma unroll
    for (int m = 16; m >= 1; m >>= 1)
        s += __shfl_xor(s, m, 32);
    if (lane == 0)
        out[e] = 1.0f / (1.0f + expf(-s));
}

// ---------------------------------------------------------------------------
extern "C" void kernel_launch(void* const* d_in, const int* in_sizes, int n_in,
                              void* d_out, int out_size, void* d_ws, size_t ws_size,
                              hipStream_t stream) {
    const float* feat     = (const float*)d_in[0];
    const int*   edge_src = (const int*)d_in[1];
    const int*   edge_dst = (const int*)d_in[2];
    const int*   src      = (const int*)d_in[3];
    const int*   dst      = (const int*)d_in[4];
    const float* W_self0  = (const float*)d_in[5];
    const float* W_neigh0 = (const float*)d_in[6];
    const float* b0       = (const float*)d_in[7];
    const float* W_self1  = (const float*)d_in[8];
    const float* W_neigh1 = (const float*)d_in[9];
    const float* b1       = (const float*)d_in[10];

    const int N = in_sizes[0] / D;     // 40000
    const int E = in_sizes[1];         // 640000
    float* scores = (float*)d_out;

    // Workspace layout
    float* deg = (float*)d_ws;                       // N (padded to 40960)
    float* agg = deg + 40960;                        // N*D  (reused for both layers)
    float* h1  = agg + (size_t)N * D;                // N*D
    float* h2  = h1  + (size_t)N * D;                // N*D

    const int THREADS = 256;
    const int edgeBlocks      = (E + THREADS - 1) / THREADS;
    const int edgeWaveBlocks  = (E * 32 + THREADS - 1) / THREADS;   // 1 wave / edge
    const int rowBlocks       = N / (16 * ROW_TILES);               // 625

    // ---- degree (shared by both layers) ----
    hipMemsetAsync(deg, 0, (size_t)N * sizeof(float), stream);
    deg_kernel<<<edgeBlocks, THREADS, 0, stream>>>(edge_dst, deg, E);

    // ---- layer 1 ----
    hipMemsetAsync(agg, 0, (size_t)N * D * sizeof(float), stream);
    scatter_kernel<<<edgeWaveBlocks, THREADS, 0, stream>>>(feat, edge_src, edge_dst, agg, E);
    sage_layer_kernel<<<rowBlocks, THREADS, 0, stream>>>(
        feat, agg, deg, W_self0, W_neigh0, b0, h1, /*relu=*/1);

    // ---- layer 2 ----
    hipMemsetAsync(agg, 0, (size_t)N * D * sizeof(float), stream);
    scatter_kernel<<<edgeWaveBlocks, THREADS, 0, stream>>>(h1, edge_src, edge_dst, agg, E);
    sage_layer_kernel<<<rowBlocks, THREADS, 0, stream>>>(
        h1, agg, deg, W_self1, W_neigh1, b1, h2, /*relu=*/0);

    // ---- edge scores ----
    edge_score_kernel<<<edgeWaveBlocks, THREADS, 0, stream>>>(h2, src, dst, scores, E);
}